// GridSamplePScan_38895223832611
// MI455X (gfx1250) — compile-verified
//
#include <hip/hip_runtime.h>
#include <stdint.h>
#include <stddef.h>

// ---------------------------------------------------------------------------
// GridSamplePScan on MI455X (gfx1250): bandwidth-bound log-step scan.
// CDNA5 paths used: wave32, global_load_async_to_lds_b128 (ASYNCcnt),
// s_wait_asynccnt, LDS ds_load for the staged streaming operand.
// ---------------------------------------------------------------------------

#define LDS_AS __attribute__((address_space(3)))
#define GLB_AS __attribute__((address_space(1)))

typedef int v4i_t __attribute__((vector_size(16)));   // 16B payload for b128

#if defined(__has_builtin)
#  if __has_builtin(__builtin_amdgcn_global_load_async_to_lds_b128)
#    define HAVE_ASYNC_LDS 1
#  endif
#endif
#ifndef HAVE_ASYNC_LDS
#  define HAVE_ASYNC_LDS 0
#endif

namespace {
constexpr int Bb = 4;
constexpr int Ll = 32;
constexpr int Cc = 16;
constexpr int Hh = 192;
constexpr int Ww = 192;
constexpr int HW = Hh * Ww;            // 36864
constexpr int PIX = 256;               // pixels per block
constexpr int TILES = HW / PIX;        // 144
static_assert(HW % PIX == 0, "tile must divide plane");

constexpr size_t IMG_ELEMS  = (size_t)Bb * Ll * Cc * HW;  // 75,497,472
constexpr size_t FLOW_ELEMS = (size_t)Bb * Ll * 2  * HW;  // 9,437,184
}  // namespace

__device__ __forceinline__ void wait_async_lds() {
#if defined(__has_builtin) && __has_builtin(__builtin_amdgcn_s_wait_asynccnt)
  __builtin_amdgcn_s_wait_asynccnt(0);
#else
  asm volatile("s_wait_asynccnt 0" ::: "memory");
#endif
}

// One log-step of the scan.  For j >= step:
//   nf[j] = fc[j] + bilinear(fp[j-step], grid(fc[j]))
//   ni[j] = ic[j] + bilinear(ip[j-step], grid(fc[j]))
// For j < step: passthrough copy so every buffer holds full state.
__global__ __launch_bounds__(PIX)
void pscan_step_kernel(const float* __restrict__ flow_src,
                       const float* __restrict__ img_src,
                       float* __restrict__ flow_dst,
                       float* __restrict__ img_dst,
                       int step, int write_flows)
{
  __shared__ float lds_ic[Cc * PIX];   // 16 KB: staged streaming ic tile

  const int t    = threadIdx.x;
  const int tile = blockIdx.x;
  const int j    = blockIdx.y;
  const int b    = blockIdx.z;
  const int p0   = tile * PIX;
  const int p    = p0 + t;

  const size_t slice_img  = (size_t)(b * Ll + j) * Cc * HW;
  const size_t slice_flow = (size_t)(b * Ll + j) * 2  * HW;

  if (j < step) {  // whole block uniform: plain passthrough copy
#pragma unroll
    for (int c = 0; c < Cc; ++c)
      img_dst[slice_img + (size_t)c * HW + p] =
          img_src[slice_img + (size_t)c * HW + p];
    if (write_flows) {
      flow_dst[slice_flow + p]      = flow_src[slice_flow + p];
      flow_dst[slice_flow + HW + p] = flow_src[slice_flow + HW + p];
    }
    return;
  }

  const size_t src_img  = (size_t)(b * Ll + (j - step)) * Cc * HW;
  const size_t src_flow = (size_t)(b * Ll + (j - step)) * 2  * HW;

#if HAVE_ASYNC_LDS
  // Kick off async staging of the 16x256 fp32 ic tile into LDS now; it
  // resolves while we do the grid math and the 72 gather loads below.
  // Writer mapping: thread t -> lane li = t&63 within channel group cg = t>>6,
  // each b128 moves 4 consecutive pixels of channel c = k*4 + cg.
  {
    GLB_AS char* gbase =
        (GLB_AS char*)(img_src + slice_img + (size_t)p0);
    LDS_AS char* lbase = (LDS_AS char*)lds_ic;
    const int li = t & 63;
    const int cg = t >> 6;
#pragma unroll
    for (int k = 0; k < 4; ++k) {
      const int c = k * 4 + cg;
      __builtin_amdgcn_global_load_async_to_lds_b128(
          (GLB_AS v4i_t*)(gbase + ((size_t)c * HW + (size_t)li * 4) * sizeof(float)),
          (LDS_AS v4i_t*)(lbase + ((size_t)c * PIX + (size_t)li * 4) * sizeof(float)),
          0, 0);
    }
  }
#endif

  // ---- warp grid (matches _warp_grid / _bilinear_sample, wrap_x=True) ----
  const int x = p % Ww;
  const int y = p / Ww;

  const float fcx = flow_src[slice_flow + p];
  const float fcy = flow_src[slice_flow + HW + p];

  const float gx = ((float)x + 0.5f) * (2.0f / (float)Ww) - 1.0f;
  const float gy = ((float)y + 0.5f) * (2.0f / (float)Hh) - 1.0f;

  float fx = gx + fcx;
  {  // fx = remainder(fx + 1, 2) - 1   (result in [0,2) like jnp.remainder)
    float a = fx + 1.0f;
    a = a - 2.0f * floorf(a * 0.5f);
    fx = a - 1.0f;
  }
  const float fy = gy + fcy;

  const float ix = (fx + 1.0f) * 0.5f * (float)Ww - 0.5f;
  const float iy = (fy + 1.0f) * 0.5f * (float)Hh - 0.5f;
  const float x0f = floorf(ix), y0f = floorf(iy);
  const float wx1 = ix - x0f,  wy1 = iy - y0f;
  const float wx0 = 1.0f - wx1, wy0 = 1.0f - wy1;

  const int x0 = (int)x0f, y0i = (int)y0f;
  const int x1 = x0 + 1,   y1i = y0i + 1;

  const float mx0 = (x0  >= 0 && x0  < Ww) ? 1.0f : 0.0f;
  const float mx1 = (x1  >= 0 && x1  < Ww) ? 1.0f : 0.0f;
  const float my0 = (y0i >= 0 && y0i < Hh) ? 1.0f : 0.0f;
  const float my1 = (y1i >= 0 && y1i < Hh) ? 1.0f : 0.0f;

  // fold zero-padding masks into the bilinear weights once for all 18 planes
  const float w00 = wy0 * wx0 * my0 * mx0;
  const float w01 = wy0 * wx1 * my0 * mx1;
  const float w10 = wy1 * wx0 * my1 * mx0;
  const float w11 = wy1 * wx1 * my1 * mx1;

  // clamped (always in-bounds) gather offsets; masked weights zero them out
  const int xc0 = min(max(x0,  0), Ww - 1);
  const int xc1 = min(max(x1,  0), Ww - 1);
  const int yc0 = min(max(y0i, 0), Hh - 1);
  const int yc1 = min(max(y1i, 0), Hh - 1);
  const int o00 = yc0 * Ww + xc0;
  const int o01 = yc0 * Ww + xc1;
  const int o10 = yc1 * Ww + xc0;
  const int o11 = yc1 * Ww + xc1;

  // ---- flows: nf = fc + sample(fp) ----
  if (write_flows) {
#pragma unroll
    for (int c = 0; c < 2; ++c) {
      const float* __restrict__ pl = flow_src + src_flow + (size_t)c * HW;
      const float s = pl[o00] * w00 + pl[o01] * w01 +
                      pl[o10] * w10 + pl[o11] * w11;
      const float fcv = (c == 0) ? fcx : fcy;
      flow_dst[slice_flow + (size_t)c * HW + p] = fcv + s;
    }
  }

  // ---- images: gather-accumulate 16 channels (overlaps async ic staging) ----
  float acc[Cc];
#pragma unroll
  for (int c = 0; c < Cc; ++c) {
    const float* __restrict__ pl = img_src + src_img + (size_t)c * HW;
    acc[c] = pl[o00] * w00 + pl[o01] * w01 +
             pl[o10] * w10 + pl[o11] * w11;
  }

#if HAVE_ASYNC_LDS
  wait_async_lds();      // our wave's async loads landed in LDS
  __syncthreads();       // cross-wave visibility of the staged tile
#pragma unroll
  for (int c = 0; c < Cc; ++c)
    img_dst[slice_img + (size_t)c * HW + p] =
        acc[c] + lds_ic[c * PIX + t];
#else
#pragma unroll
  for (int c = 0; c < Cc; ++c)
    img_dst[slice_img + (size_t)c * HW + p] =
        acc[c] + img_src[slice_img + (size_t)c * HW + p];
#endif
}

extern "C" void kernel_launch(void* const* d_in, const int* in_sizes, int n_in,
                              void* d_out, int out_size, void* d_ws, size_t ws_size,
                              hipStream_t stream) {
  (void)in_sizes; (void)n_in; (void)out_size; (void)ws_size;

  const float* flows0  = (const float*)d_in[0];  // [B,L,2,H,W]
  const float* images0 = (const float*)d_in[1];  // [B,L,C,H,W]

  // Ping-pong buffers. Images alternate between d_out and ws; parity of the
  // 5 steps (1,2,4,8,16) is arranged so the final result lands in d_out.
  float* imgA  = (float*)d_out;                  // image ping (final)
  float* imgB  = (float*)d_ws;                   // image pong
  float* flowA = imgB + IMG_ELEMS;               // flow ping
  float* flowB = flowA + FLOW_ELEMS;             // flow pong
  // ws usage: (IMG_ELEMS + 2*FLOW_ELEMS) * 4B  ~= 377.5 MB

  dim3 grid(TILES, Ll, Bb);
  dim3 blk(PIX);

  //            flow_src  img_src   flow_dst  img_dst  step  write_flows
  pscan_step_kernel<<<grid, blk, 0, stream>>>(flows0, images0, flowA, imgA,  1, 1);
  pscan_step_kernel<<<grid, blk, 0, stream>>>(flowA,  imgA,    flowB, imgB,  2, 1);
  pscan_step_kernel<<<grid, blk, 0, stream>>>(flowB,  imgB,    flowA, imgA,  4, 1);
  pscan_step_kernel<<<grid, blk, 0, stream>>>(flowA,  imgA,    flowB, imgB,  8, 1);
  pscan_step_kernel<<<grid, blk, 0, stream>>>(flowB,  imgB,    flowA, imgA, 16, 0);
}